// RetentionMechanism_89524298318642
// MI455X (gfx1250) — compile-verified
//
#include <hip/hip_runtime.h>
#include <cstdint>
#include <cstddef>

typedef __attribute__((ext_vector_type(2))) float        v2f;
typedef __attribute__((ext_vector_type(8))) float        v8f;
typedef __attribute__((ext_vector_type(4))) unsigned int v4u;
typedef __attribute__((ext_vector_type(8))) int          v8i;
typedef __attribute__((ext_vector_type(4))) int          v4i;

#define B_SZ   8
#define N_SZ   4096
#define D_SZ   2048
#define CHUNK  512            // timesteps handled per workgroup
#define WARM   32             // warm-up steps (0.5^32 ~ 2e-10, below fp32 ulp)
#define TSTAGE 32             // timestep rows per TDM stage
#define WGCH   128            // channels per workgroup (8 waves x 16)
#define ROWPAD 132            // floats per LDS row: 128 data + 4 TDM pad
#define STAGE_FLOATS (TSTAGE * ROWPAD)
#define STAGE_BYTES  (STAGE_FLOATS * 4)

// Compiler-only fence: combined with the escaped LDS pointer below, this makes
// the optimizer assume the (TDM-written) LDS buffer may have been stored to,
// so its loads are neither folded to poison nor LICM-hoisted out of the loop.
__device__ __forceinline__ void lds_compiler_fence() {
    asm volatile("" ::: "memory");
}

// One TDM descriptor moves a [TSTAGE x 128] fp32 tile (row stride D_SZ) from
// global into LDS, inserting 4 pad DWORDs after every 128 stored DWORDs so
// LDS rows are 132 floats (bank-conflict stagger). D# layout per CDNA5 ISA §8.
// 6-arg builtin on this toolchain: (g0 v4u, g1 v8i, g2 v4i, g3 v4i, g4 v8i, cpol).
__device__ __forceinline__ void tdm_stage_load(const float* gsrc, unsigned lds_off)
{
    unsigned long long ga = (unsigned long long)gsrc;
    v4u g0;
    g0[0] = 1u;                                                   // count=1, user D#
    g0[1] = lds_off;                                              // lds_addr (bytes)
    g0[2] = (unsigned)(ga & 0xFFFFFFFFull);                       // global_addr[31:0]
    g0[3] = (unsigned)((ga >> 32) & 0x01FFFFFFull) | (2u << 30);  // addr[56:32] | type=2
    v8i g1;
    g1[0] = (2 << 16)            // data_size = 4B
          | (1 << 20)            // pad_enable
          | (6 << 22)            // pad_interval: 128 DWORDs
          | (3 << 25);           // pad_amount: 4 DWORDs
    g1[1] = (int)0xFFFF0000u;    // barrier_addr=0 | tensor_dim0[15:0] (huge: no OOB clamp)
    g1[2] = (int)0xFFFF7FFFu;    // tensor_dim0[31:16] | tensor_dim1[15:0]
    g1[3] = (int)0x00807FFFu;    // tensor_dim1[31:16] | tile_dim0 = 128
    g1[4] = TSTAGE;              // tile_dim1 = 32, tile_dim2 = 0 (2D)
    g1[5] = D_SZ;                // tensor_dim0_stride = 2048 elements
    g1[6] = 0;
    g1[7] = 0;
    v4i gz;  gz[0] = 0; gz[1] = 0; gz[2] = 0; gz[3] = 0;
    v8i gz8; gz8[0] = 0; gz8[1] = 0; gz8[2] = 0; gz8[3] = 0;
             gz8[4] = 0; gz8[5] = 0; gz8[6] = 0; gz8[7] = 0;
    __builtin_amdgcn_tensor_load_to_lds(g0, g1, gz, gz, gz8, 0);
}

__global__ __launch_bounds__(256)
void retention_wmma_kernel(const float* __restrict__ x, float* __restrict__ out)
{
    __shared__ float lds[2 * STAGE_FLOATS];

    // Escape the LDS array's address into an asm operand: without this, the
    // module contains no stores to `lds` (the TDM only sees its integer byte
    // offset), globals-AA proves it "never written", and every read of it is
    // folded to poison (seen in rounds 2 and 3: constant B operand, ds=5).
    // After escaping, asm blocks with "memory" clobbers may store through it.
    {
        const float* esc = &lds[0];
        asm volatile("" :: "v"(esc) : "memory");
    }

    const int lane   = threadIdx.x & 31;
    const int wv     = threadIdx.x >> 5;    // wave 0..7, each owns 16 channels
    const int m      = lane & 15;           // matrix row (A) / column (B,C,D)
    const int hi     = lane >> 4;           // half-wave select
    const int khi    = hi * 2;              // K offset within 4-chunk for A/B frags
    const int rowoff = hi * 8;              // C/D rows: M=j (lo half), M=j+8 (hi half)

    // Scalar (SGPR) wave id -> real s_cbranch guard. TENSOR_LOAD_TO_LDS ignores
    // EXEC, so an exec-masked guard would make all 8 waves issue the DMA (8x
    // traffic). A scalar branch guarantees exactly one issue per workgroup.
    const int wv_s = __builtin_amdgcn_readfirstlane((int)threadIdx.x) >> 5;

    const int dblk  = blockIdx.x;           // 0..15  (channel block of 128)
    const int chunk = blockIdx.y;           // 0..7   (512-step chunk)
    const int b     = blockIdx.z;           // 0..7

    const int d0   = dblk * WGCH;
    const int t0   = chunk * CHUNK;
    const int tbeg = (chunk == 0) ? 0 : (t0 - WARM);
    const int nstages = (chunk == 0) ? (CHUNK / TSTAGE) : ((CHUNK + WARM) / TSTAGE);

    // A = L, lower-triangular decay: L[m][k] = 2^(k-m) for k<=m (exact powers of 2).
    // 16x4 f32 A-frag layout: lanes0-15 rows m, VGPR0=K{0|2}, VGPR1=K{1|3}.
    v2f afrag[4];
#pragma unroll
    for (int s = 0; s < 4; ++s) {
        int k0 = 4 * s + khi;
        afrag[s].x = (k0     <= m) ? __builtin_ldexpf(1.0f, k0 - m)     : 0.0f;
        afrag[s].y = (k0 + 1 <= m) ? __builtin_ldexpf(1.0f, k0 + 1 - m) : 0.0f;
    }
    // Carry weights: xi[t0+M] += 0.5^(M+1) * carry, M = j + rowoff.
    float wcar[8];
#pragma unroll
    for (int j = 0; j < 8; ++j)
        wcar[j] = __builtin_ldexpf(1.0f, -(j + 1 + rowoff));

    const float* gx = x + (size_t)b * N_SZ * D_SZ + d0;
    const int ch = wv * 16 + m;             // channel column within the 128-wide block

    // Prime double-buffered TDM pipeline (wave 0 drives the DMA engine).
    if (wv_s == 0) {
        tdm_stage_load(gx + (size_t)tbeg * D_SZ, 0u);
        if (nstages > 1)
            tdm_stage_load(gx + (size_t)(tbeg + TSTAGE) * D_SZ, STAGE_BYTES);
    }

    float carry = 0.0f;

    for (int s = 0; s < nstages; ++s) {
        if (wv_s == 0) {                     // stage s complete (TDM in-order per wave)
            if (s + 1 < nstages) __builtin_amdgcn_s_wait_tensorcnt((short)1);
            else                 __builtin_amdgcn_s_wait_tensorcnt((short)0);
        }
        __syncthreads();
        lds_compiler_fence();                // TDM wrote LDS: force in-loop reloads

        const float* buf    = &lds[(s & 1) * STAGE_FLOATS];
        const int    tstage = tbeg + s * TSTAGE;

#pragma unroll
        for (int tile = 0; tile < TSTAGE / 16; ++tile) {
            const int tg0 = tstage + tile * 16;

            // C preload = carry * decay column (folds the recurrence carry-in).
            v8f acc;
#pragma unroll
            for (int j = 0; j < 8; ++j) acc[j] = carry * wcar[j];

            // xi_tile = L @ b_tile + C via 4 chained K=4 fp32 WMMAs.
#pragma unroll
            for (int s4 = 0; s4 < 4; ++s4) {
                int ta = tile * 16 + 4 * s4 + khi;
                float xa = buf[ta * ROWPAD + ch];
                float xb = buf[(ta + 1) * ROWPAD + ch];
                v2f bb;
                bb.x = 0.5f * __builtin_amdgcn_rcpf(1.0f + __expf(-xa));
                bb.y = 0.5f * __builtin_amdgcn_rcpf(1.0f + __expf(-xb));
                acc = __builtin_amdgcn_wmma_f32_16x16x4_f32(
                        false, afrag[s4], false, bb, (short)0, acc, false, false);
            }

            if (tg0 >= t0) {                 // skip warm-up rows
                float* go = out + ((size_t)b * N_SZ + tg0 + rowoff) * D_SZ + d0 + ch;
#pragma unroll
                for (int j = 0; j < 8; ++j)  // write-once stream: bypass-cache stores
                    __builtin_nontemporal_store(acc[j], go + (size_t)j * D_SZ);
            }

            // New carry = xi row 15 (VGPR7, lanes 16-31) broadcast per column.
            carry = __shfl(acc[7], 16 + m, 32);
        }

        __syncthreads();                     // everyone done reading buf (s&1)
        if (wv_s == 0 && s + 2 < nstages)
            tdm_stage_load(gx + (size_t)(tbeg + (s + 2) * TSTAGE) * D_SZ,
                           (unsigned)((s & 1) * STAGE_BYTES));
    }
}

extern "C" void kernel_launch(void* const* d_in, const int* in_sizes, int n_in,
                              void* d_out, int out_size, void* d_ws, size_t ws_size,
                              hipStream_t stream)
{
    (void)in_sizes; (void)n_in; (void)out_size; (void)d_ws; (void)ws_size;
    const float* x   = (const float*)d_in[0];
    float*       out = (float*)d_out;
    dim3 grid(D_SZ / WGCH, N_SZ / CHUNK, B_SZ);   // 16 x 8 x 8 = 1024 WGs, 8 waves each
    retention_wmma_kernel<<<grid, 256, 0, stream>>>(x, out);
}